// StraightThroughQuantizer_58488864637399
// MI455X (gfx1250) — compile-verified
//
#include <hip/hip_runtime.h>

typedef float v2f __attribute__((ext_vector_type(2)));
typedef float v4f __attribute__((ext_vector_type(4)));
typedef float v8f __attribute__((ext_vector_type(8)));

#define H_DIM       128
#define K_CODES     4096
#define N_ROWS      32768
#define TILE_N      16
#define NUM_TILES   (K_CODES / TILE_N)        // 256
#define FRAG_STRIDE 68                        // 64 data floats + 16B pad per frag-row
#define LDS_TILE    (2 * TILE_N * FRAG_STRIDE)// 32 frag-rows per tile
#define WAVES_WG    8
#define ROWS_WAVE   16
#define ROWS_WG     (WAVES_WG * ROWS_WAVE)    // 128
#define NUM_WGS     (N_ROWS / ROWS_WG)        // 256
#define NUM_PART    (NUM_WGS * WAVES_WG)      // 2048 loss partials

// ---------------- kernel 1: ||w||^2 per code ----------------
__global__ void vq_wnorm_kernel(const float* __restrict__ W,
                                float* __restrict__ wn) {
    int k = blockIdx.x * blockDim.x + threadIdx.x;
    if (k >= K_CODES) return;
    const float* row = W + (size_t)k * H_DIM;
    float s = 0.f;
#pragma unroll 8
    for (int d = 0; d < H_DIM; ++d) { float v = row[d]; s = fmaf(v, v, s); }
    wn[k] = s;
}

// ---------------- kernel 2: main WMMA distance + argmin + STE ----------------
__global__ __launch_bounds__(256, 1)
void vq_main_kernel(const float* __restrict__ Z,
                    const float* __restrict__ W,
                    const float* __restrict__ wn,
                    float* __restrict__ outQ,
                    float* __restrict__ outIdx,
                    float* __restrict__ partials) {
    // fragment-major codebook tile, double buffered:
    //   frag-row f = 2*code + hf holds that code's K-pairs for half hf,
    //   element j at float offset 2j within the row (row stride 68 floats)
    __shared__ float ldsB[2 * LDS_TILE];

    const int tid  = threadIdx.x;
    const int wave = tid >> 5;
    const int lane = tid & 31;
    const int hf   = lane >> 4;       // half of wave: selects K pair
    const int slot = lane & 15;       // row (A) / column (B) within tile

    const int rowBase = blockIdx.x * ROWS_WG + wave * ROWS_WAVE;

    // A fragments: ISA 16x4 f32 layout, lane(hf,slot) pair j holds
    //   Z[rowBase+slot][4j + 2*hf + {0,1}]
    v2f a[32];
    {
        const float* zrow = Z + (size_t)(rowBase + slot) * H_DIM;
#pragma unroll
        for (int j = 0; j < 32; ++j)
            a[j] = *(const v2f*)(zrow + 4 * j + 2 * hf);
    }

    float bestVal[8];
    int   bestIdx[8];
#pragma unroll
    for (int r = 0; r < 8; ++r) { bestVal[r] = 3.4e38f; bestIdx[r] = 0; }

    // staging map: thread -> (code c, 8-dim segment seg) of the 16x128 tile
    const int c   = tid >> 4;          // 0..15
    const int seg = tid & 15;          // 0..15
    const float* gsrc = W + (size_t)c * H_DIM + seg * 8;
    float* dst0 = ldsB + (c * 2 + 0) * FRAG_STRIDE + 4 * seg;   // hf=0 row
    float* dst1 = ldsB + (c * 2 + 1) * FRAG_STRIDE + 4 * seg;   // hf=1 row

    // prologue: fetch tile 0 into registers
    v4f g0 = *(const v4f*)(gsrc);
    v4f g1 = *(const v4f*)(gsrc + 4);

    for (int t = 0; t < NUM_TILES; ++t) {
        const int boff = (t & 1) * LDS_TILE;
        // commit prefetched tile t into LDS buffer t&1 (fragment-major)
        *(v4f*)(dst0 + boff) = __builtin_shufflevector(g0, g1, 0, 1, 4, 5);
        *(v4f*)(dst1 + boff) = __builtin_shufflevector(g0, g1, 2, 3, 6, 7);
        __syncthreads();   // single barrier/tile: reads drained before signal

        // issue global fetch of tile t+1; latency covered by 16 WMMAs below
        if (t + 1 < NUM_TILES) {
            const float* nsrc = gsrc + (size_t)(t + 1) * TILE_N * H_DIM;
            g0 = *(const v4f*)(nsrc);
            g1 = *(const v4f*)(nsrc + 4);
        }

        // 16x16 dot tile: 8 x ds_load_b128, each feeding 2 WMMAs (1 per acc)
        const float* bbase = ldsB + boff + (slot * 2 + hf) * FRAG_STRIDE;
        v8f acc0 = {}, acc1 = {};
#pragma unroll
        for (int q = 0; q < 8; ++q) {
            v4f bq = *(const v4f*)(bbase + 4 * q);        // frags j=2q, 2q+1
            v2f blo = __builtin_shufflevector(bq, bq, 0, 1);
            v2f bhi = __builtin_shufflevector(bq, bq, 2, 3);
            acc0 = __builtin_amdgcn_wmma_f32_16x16x4_f32(
                       false, a[2 * q],     false, blo, (short)0, acc0, false, false);
            acc1 = __builtin_amdgcn_wmma_f32_16x16x4_f32(
                       false, a[2 * q + 1], false, bhi, (short)0, acc1, false, false);
        }

        // argmin score = ||w||^2 - 2*dot  (||z||^2 is row-constant)
        const int codeBase = t * TILE_N;
        const float w2 = wn[codeBase + slot];
        const int   n  = codeBase + slot;
#pragma unroll
        for (int r = 0; r < 8; ++r) {
            float s = fmaf(-2.0f, acc0[r] + acc1[r], w2);
            if (s < bestVal[r]) { bestVal[r] = s; bestIdx[r] = n; }
        }
    }

    // reduce (val,idx) across the 16 column slots in each wave half;
    // ties -> smaller index (argmin first-occurrence semantics)
#pragma unroll
    for (int m = 1; m <= 8; m <<= 1) {
#pragma unroll
        for (int r = 0; r < 8; ++r) {
            float ov = __shfl_xor(bestVal[r], m, 32);
            int   oi = __shfl_xor(bestIdx[r], m, 32);
            if (ov < bestVal[r] || (ov == bestVal[r] && oi < bestIdx[r])) {
                bestVal[r] = ov; bestIdx[r] = oi;
            }
        }
    }
    // lanes 0-15 now hold rows rowBase+0..7; lanes 16-31 rows rowBase+8..15
    if (slot == 0) {
#pragma unroll
        for (int r = 0; r < 8; ++r)
            outIdx[rowBase + hf * 8 + r] = (float)bestIdx[r];
    }

    // epilogue: gather zq, straight-through output, loss partial
    float lsum = 0.f;
#pragma unroll
    for (int r = 0; r < 16; ++r) {
        const int src = (r < 8) ? 0 : 16;
        const int ci  = __shfl(bestIdx[r & 7], src, 32);
        const float* wrow = W + (size_t)ci * H_DIM;
        const float* zrow = Z + (size_t)(rowBase + r) * H_DIM;
        float*       orow = outQ + (size_t)(rowBase + r) * H_DIM;
        v4f wv = *(const v4f*)(wrow + lane * 4);
        v4f zv = *(const v4f*)(zrow + lane * 4);
        v4f d;
#pragma unroll
        for (int e = 0; e < 4; ++e) {
            float diff = wv[e] - zv[e];
            d[e] = zv[e] + diff;              // z + sg(zq - z)
            lsum = fmaf(diff, diff, lsum);
        }
        *(v4f*)(orow + lane * 4) = d;
    }
#pragma unroll
    for (int m = 1; m < 32; m <<= 1) lsum += __shfl_xor(lsum, m, 32);
    if (lane == 0) partials[blockIdx.x * WAVES_WG + wave] = lsum;
}

// ---------------- kernel 3: deterministic loss finalize ----------------
__global__ void vq_finalize_kernel(const float* __restrict__ partials,
                                   float* __restrict__ lossOut) {
    __shared__ float red[256];
    const int t = threadIdx.x;
    float s = 0.f;
#pragma unroll
    for (int i = 0; i < NUM_PART / 256; ++i) s += partials[t * (NUM_PART / 256) + i];
    red[t] = s;
    __syncthreads();
    if (t == 0) {
        float tot = 0.f;
        for (int i = 0; i < 256; ++i) tot += red[i];
        // loss = 0.25*commit + 1.0*quant, both forward-equal to mean((zq-z)^2)
        *lossOut = 1.25f * tot / (float)((size_t)N_ROWS * H_DIM);
    }
}

extern "C" void kernel_launch(void* const* d_in, const int* in_sizes, int n_in,
                              void* d_out, int out_size, void* d_ws, size_t ws_size,
                              hipStream_t stream) {
    const float* Z = (const float*)d_in[0];            // (8,4096,128) f32
    const float* W = (const float*)d_in[1];            // (4096,128)  f32

    float* out     = (float*)d_out;
    float* outQ    = out;                              // N*H quantized_z
    float* outIdx  = outQ + (size_t)N_ROWS * H_DIM;    // N indices (as float)
    float* lossOut = outIdx + N_ROWS;                  // 1 loss scalar

    float* ws       = (float*)d_ws;
    float* wn       = ws;                              // [0, 4096)   ||w||^2
    float* partials = ws + K_CODES;                    // [4096, 6144) loss partials

    hipLaunchKernelGGL(vq_wnorm_kernel, dim3(K_CODES / 256), dim3(256), 0, stream,
                       W, wn);
    hipLaunchKernelGGL(vq_main_kernel, dim3(NUM_WGS), dim3(256), 0, stream,
                       Z, W, wn, outQ, outIdx, partials);
    hipLaunchKernelGGL(vq_finalize_kernel, dim3(1), dim3(256), 0, stream,
                       partials, lossOut);
}